// PhyloNeighbours_56040733278560
// MI455X (gfx1250) — compile-verified
//
#include <hip/hip_runtime.h>

#define NC 512      // nb_coordinates
#define NF 2000     // nb_features
#define NFILT 16    // filters
#define NB 64       // batch
#define KNN 8

typedef __attribute__((ext_vector_type(2))) float v2f;
typedef __attribute__((ext_vector_type(4))) float v4f;
typedef __attribute__((ext_vector_type(8))) float v8f;

#define FMAXV 3.402823466e+38f

// coordsT layout: [f][cgrp][i][q], cgrp = kb*2 + half (128 groups), q in 0..3
//   element (f, cgrp, i, q) = coords[c][i][f] with
//   c = (cgrp>>1)*8 + (cgrp&1)*2 + ((q>>1)*4 + (q&1))
// One b128 load at (f, kb*2+half, i) yields the lane's A/B K-values for TWO
// consecutive 16x16x4 WMMA steps: (k0, k0+1) in .xy and (k0+4, k0+5) in .zw.
#define FSTRIDE (128L * NF * 4)   // floats per filter = 1,024,000

// ---------------------------------------------------------------------------
// Kernel T: transpose + K-quad interleave.  grid = 128 cgrp x 125 i-tiles.
// ---------------------------------------------------------------------------
__global__ __launch_bounds__(256)
void transpose_coords(const float* __restrict__ coords, float* __restrict__ coordsT) {
    __shared__ float tile[4][16][17];
    const int blk   = blockIdx.x;            // 128 * 125 = 16000 blocks
    const int cgrp  = blk / 125;
    const int i0    = (blk - cgrp * 125) * 16;
    const int cbase = (cgrp >> 1) * 8 + (cgrp & 1) * 2;

    const int t   = threadIdx.x;
    const int f_l = t & 15;
    const int r_l = t >> 4;                  // 0..15
    const int coff[4] = {0, 1, 4, 5};
#pragma unroll
    for (int j = 0; j < 4; ++j) {            // 64B-coalesced reads per 16 lanes
        tile[j][r_l][f_l] =
            coords[(long)(cbase + coff[j]) * (NF * NFILT) + (i0 + r_l) * NFILT + f_l];
    }
    __syncthreads();

    const int f_s = t >> 4;
    const int x   = t & 15;
    const long obase = (long)f_s * FSTRIDE + (long)cgrp * (NF * 4) + (long)i0 * 4;
#pragma unroll
    for (int it = 0; it < 4; ++it) {
        int pos = x + 16 * it;               // 0..63 -> i = pos>>2, q = pos&3
        coordsT[obase + pos] = tile[pos & 3][pos >> 2][f_s];  // contiguous stores
    }
}

// ---------------------------------------------------------------------------
// Kernel S: per-filter squared column norms, b128 walks of the quad layout.
// ---------------------------------------------------------------------------
__global__ __launch_bounds__(256)
void compute_sq(const float* __restrict__ coordsT, float* __restrict__ sq) {
    int t = blockIdx.x * 256 + threadIdx.x;  // exactly NFILT*NF = 32000 threads
    int f = t / NF;
    int i = t - f * NF;
    const v4f* p = (const v4f*)(coordsT + (long)f * FSTRIDE + (long)i * 4);
    float s = 0.f;
    for (int g = 0; g < 128; ++g) {          // all 128 cgrp blocks
        v4f v = *p;
        s += v.x * v.x + v.y * v.y + v.z * v.z + v.w * v.w;
        p += NF;                              // advance one cgrp (8000 floats)
    }
    sq[t] = s;
}

// ---------------------------------------------------------------------------
// Kernel 1: fused Gram-GEMM (fp32 WMMA 16x16x4) + distance + top-8 per row.
// 2-way j-tile register blocking: shared A load -> 3 b128 loads per 4 WMMAs.
// ---------------------------------------------------------------------------
#define TILES_PER_PASS 32
#define NPASS 4
#define LDS_STRIDE 516

__global__ __launch_bounds__(128)
void gram_topk(const float* __restrict__ coordsT, const float* __restrict__ sq,
               int* __restrict__ idx_out) {
    __shared__ float dist_lds[16][LDS_STRIDE];
    __shared__ float cand_val[16][8][8];
    __shared__ int   cand_idx[16][8][8];

    const int bid = blockIdx.x;           // 2000 blocks
    const int f   = bid & 15;
    const int ib  = (bid >> 4) * 16;      // row tile base

    const float* __restrict__ A   = coordsT + (long)f * FSTRIDE;
    const float* __restrict__ sqf = sq + f * NF;

    const int tid  = threadIdx.x;
    const int wave = tid >> 5;
    const int lane = tid & 31;
    const int half = lane >> 4;
    const int ln   = lane & 15;

    const int selRow = tid >> 3;          // 0..15
    const int selT   = tid & 7;           // 0..7
    float best[8]; int bidx[8];
#pragma unroll
    for (int s = 0; s < 8; ++s) { best[s] = FMAXV; bidx[s] = 0; }

    float sqRow[8];
#pragma unroll
    for (int v = 0; v < 8; ++v) sqRow[v] = sqf[ib + v + 8 * half];

    const float* Ahalf  = A + (long)half * (NF * 4);
    const float* paBase = Ahalf + (long)(ib + ln) * 4;

    for (int pass = 0; pass < NPASS; ++pass) {
        const int t0 = pass * TILES_PER_PASS;
        const int t1 = (t0 + TILES_PER_PASS < 125) ? (t0 + TILES_PER_PASS) : 125;
        const int colBase = t0 * 16;
        const int nCols = (t1 - t0) * 16;

        // ---- GEMM phase: wave-uniform, 2 adjacent j-tiles per iteration ----
        for (int jt = t0 + 2 * wave; jt < t1; jt += 8) {
            const int jb = jt * 16;
            if (jt + 1 < t1) {
                // dual-tile path: one A load feeds both accumulators
                v8f acc0 = {0.f, 0.f, 0.f, 0.f, 0.f, 0.f, 0.f, 0.f};
                v8f acc1 = {0.f, 0.f, 0.f, 0.f, 0.f, 0.f, 0.f, 0.f};
                const float* pa  = paBase;
                const float* pb0 = Ahalf + (long)(jb + ln) * 4;
                const float* pb1 = pb0 + 64;                 // next 16-col tile
                for (int kb = 0; kb < NC / 8; ++kb) {
                    v4f a4 = *(const v4f*)pa;
                    v4f b4 = *(const v4f*)pb0;
                    v4f c4 = *(const v4f*)pb1;
                    v2f a0 = __builtin_shufflevector(a4, a4, 0, 1);
                    v2f a1 = __builtin_shufflevector(a4, a4, 2, 3);
                    v2f b0 = __builtin_shufflevector(b4, b4, 0, 1);
                    v2f b1 = __builtin_shufflevector(b4, b4, 2, 3);
                    v2f c0 = __builtin_shufflevector(c4, c4, 0, 1);
                    v2f c1 = __builtin_shufflevector(c4, c4, 2, 3);
                    acc0 = __builtin_amdgcn_wmma_f32_16x16x4_f32(
                            false, a0, false, b0, (short)0, acc0, false, false);
                    acc1 = __builtin_amdgcn_wmma_f32_16x16x4_f32(
                            false, a0, false, c0, (short)0, acc1, false, false);
                    acc0 = __builtin_amdgcn_wmma_f32_16x16x4_f32(
                            false, a1, false, b1, (short)0, acc0, false, false);
                    acc1 = __builtin_amdgcn_wmma_f32_16x16x4_f32(
                            false, a1, false, c1, (short)0, acc1, false, false);
                    pa  += 2 * 4 * NF;
                    pb0 += 2 * 4 * NF;
                    pb1 += 2 * 4 * NF;
                }
                const int lcol0 = jb - colBase + ln;
                const float sc0 = sqf[jb + ln];
                const float sc1 = sqf[jb + 16 + ln];
#pragma unroll
                for (int v = 0; v < 8; ++v) {
                    float d0 = fmaxf(sqRow[v] + sc0 - 2.0f * acc0[v], 0.0f);
                    float d1 = fmaxf(sqRow[v] + sc1 - 2.0f * acc1[v], 0.0f);
                    dist_lds[v + 8 * half][lcol0]      = d0;
                    dist_lds[v + 8 * half][lcol0 + 16] = d1;
                }
            } else {
                // single leftover tile (odd tail of pass 3) - wave-uniform branch
                v8f acc = {0.f, 0.f, 0.f, 0.f, 0.f, 0.f, 0.f, 0.f};
                const float* pa = paBase;
                const float* pb = Ahalf + (long)(jb + ln) * 4;
                for (int kb = 0; kb < NC / 8; ++kb) {
                    v4f a4 = *(const v4f*)pa;
                    v4f b4 = *(const v4f*)pb;
                    v2f a0 = __builtin_shufflevector(a4, a4, 0, 1);
                    v2f b0 = __builtin_shufflevector(b4, b4, 0, 1);
                    acc = __builtin_amdgcn_wmma_f32_16x16x4_f32(
                            false, a0, false, b0, (short)0, acc, false, false);
                    v2f a1 = __builtin_shufflevector(a4, a4, 2, 3);
                    v2f b1 = __builtin_shufflevector(b4, b4, 2, 3);
                    acc = __builtin_amdgcn_wmma_f32_16x16x4_f32(
                            false, a1, false, b1, (short)0, acc, false, false);
                    pa += 2 * 4 * NF;
                    pb += 2 * 4 * NF;
                }
                const int lcol = jb - colBase + ln;
                const float sc = sqf[jb + ln];
#pragma unroll
                for (int v = 0; v < 8; ++v) {
                    float d = fmaxf(sqRow[v] + sc - 2.0f * acc[v], 0.0f);
                    dist_lds[v + 8 * half][lcol] = d;
                }
            }
        }
        __syncthreads();

        // ---- selection: running sorted top-8 per (row, selT) ----
        const int nc8 = nCols >> 3;
        for (int c = 0; c < nc8; ++c) {
            const int lc = c * 8 + selT;
            const float d = dist_lds[selRow][lc];
            if (d < best[7]) {
                const int col = colBase + lc;
                int p = 7;
                while (p > 0 && d < best[p - 1]) {
                    best[p] = best[p - 1]; bidx[p] = bidx[p - 1]; --p;
                }
                best[p] = d; bidx[p] = col;
            }
        }
        __syncthreads();
    }

    // ---- merge 8 candidate lists per row into final sorted top-8 ----
#pragma unroll
    for (int s = 0; s < 8; ++s) {
        cand_val[selRow][selT][s] = best[s];
        cand_idx[selRow][selT][s] = bidx[s];
    }
    __syncthreads();
    if (tid < 16) {
        const int row = tid;
        float* cv = &cand_val[row][0][0];
        int*   ci = &cand_idx[row][0][0];
        for (int k = 0; k < KNN; ++k) {
            float mv = cv[0]; int mp = 0;
            for (int s = 1; s < 64; ++s)
                if (cv[s] < mv) { mv = cv[s]; mp = s; }
            idx_out[((ib + row) * KNN + k) * NFILT + f] = ci[mp];
            cv[mp] = FMAXV;
        }
    }
}

// ---------------------------------------------------------------------------
// Kernel 2: gather  out[b, i*8+k, c] = inputs[b, idx[i,k,c], c]
// ---------------------------------------------------------------------------
__global__ __launch_bounds__(256)
void gather_out(const float* __restrict__ inputs, const int* __restrict__ idx,
                float* __restrict__ out) {
    const long t = (long)blockIdx.x * 256 + threadIdx.x;  // < 64*16000*16
    const int  c = (int)(t & 15);
    const long rb = t >> 4;
    const int  r = (int)(rb % (NF * KNN));
    const int  b = (int)(rb / (NF * KNN));
    const int  src = idx[r * NFILT + c];
    out[t] = inputs[((long)b * NF + src) * NFILT + c];
}

// ---------------------------------------------------------------------------
extern "C" void kernel_launch(void* const* d_in, const int* in_sizes, int n_in,
                              void* d_out, int out_size, void* d_ws, size_t ws_size,
                              hipStream_t stream) {
    (void)in_sizes; (void)n_in; (void)out_size; (void)ws_size;
    const float* inputs = (const float*)d_in[0];   // (64, 2000, 16)
    const float* coords = (const float*)d_in[1];   // (512, 2000, 16)
    float* out = (float*)d_out;                    // (64, 16000, 16)

    float* coordsT = (float*)d_ws;                 // 16 * 1,024,000 floats
    float* sq      = coordsT + (long)NFILT * FSTRIDE;   // 32000 floats
    int*   idx_ws  = (int*)(sq + NFILT * NF);           // 16000*16 ints

    transpose_coords<<<128 * 125, 256, 0, stream>>>(coords, coordsT);
    compute_sq<<<(NFILT * NF) / 256, 256, 0, stream>>>(coordsT, sq);
    gram_topk<<<NFILT * (NF / 16), 128, 0, stream>>>(coordsT, sq, idx_ws);
    gather_out<<<(NB * NF * KNN * NFILT) / 256, 256, 0, stream>>>(inputs, idx_ws, out);
}